// NAA_GCN_24481313587809
// MI455X (gfx1250) — compile-verified
//
#include <hip/hip_runtime.h>

typedef __attribute__((ext_vector_type(2))) float v2f;
typedef __attribute__((ext_vector_type(8))) float v8f;

#define IN_DIM 256
#define HID    128
#define OUTD   2
#define BN_EPS 1e-5f

// ---------------- init: deg=1 (self loop), BN sums=0, sigw=sigmoid(fi) ----------------
__global__ void k_init(float* __restrict__ deg, float* __restrict__ sums,
                       float* __restrict__ sumsq, float* __restrict__ sigw,
                       const float* __restrict__ fi, int N) {
  int i = blockIdx.x * blockDim.x + threadIdx.x;
  if (i < N) deg[i] = 1.0f;                       // self-loop contributes 1
  if (i < IN_DIM) sigw[i] = 1.0f / (1.0f + __expf(-fi[i]));
  if (i < HID) { sums[i] = 0.0f; sumsq[i] = 0.0f; }
}

// ---------------- degree accumulation over edges ----------------
__global__ void k_deg(const int* __restrict__ ei, float* __restrict__ deg, int E) {
  int e = blockIdx.x * blockDim.x + threadIdx.x;
  if (e < E) atomicAdd(&deg[ei[E + e]], 1.0f);
}

// ---------------- deg -> rsqrt(deg) in place ----------------
__global__ void k_dinv(float* __restrict__ deg, int N) {
  int i = blockIdx.x * blockDim.x + threadIdx.x;
  if (i < N) {
    float d = deg[i];
    deg[i] = (d > 0.0f) ? rsqrtf(d) : 0.0f;
  }
}

// ---------------- h1 = (x * sigw) @ W1 via V_WMMA_F32_16X16X4_F32 ----------------
// block = 256 threads = 8 waves; block computes 16 rows x 128 cols.
// wave w computes the 16x16 tile at columns [16w, 16w+16).
#define XS_STRIDE 260
__global__ __launch_bounds__(256) void k_gemm1(const float* __restrict__ x,
                                               const float* __restrict__ W1,
                                               const float* __restrict__ sigw,
                                               float* __restrict__ h1, int N) {
  __shared__ float xs[16][XS_STRIDE];
  const int m0 = blockIdx.x * 16;
  const int t  = threadIdx.x;           // 0..255 == column of x tile
  const float sw = sigw[t];
  for (int i = 0; i < 16; ++i) {
    int row = m0 + i;
    xs[i][t] = (row < N) ? x[row * IN_DIM + t] * sw : 0.0f;
  }
  __syncthreads();

  const int lane = t & 31;
  const int wv   = t >> 5;              // 0..7 -> output column tile
  const int half = lane >> 4;           // 0: K pair {0,1}; 1: K pair {2,3}
  const int m    = lane & 15;           // A row index / B col index
  const int n0   = wv * 16;

  v8f acc = {0.f, 0.f, 0.f, 0.f, 0.f, 0.f, 0.f, 0.f};
  for (int k0 = 0; k0 < IN_DIM; k0 += 4) {
    const int kk = k0 + (half << 1);
    v2f a, b;
    // A 16x4 f32: VGPR0 = K{0|2}, VGPR1 = K{1|3}, M = lane%16
    a[0] = xs[m][kk];
    a[1] = xs[m][kk + 1];
    // B 4x16 f32: same K striping, N = lane%16
    b[0] = W1[kk * HID + n0 + m];
    b[1] = W1[(kk + 1) * HID + n0 + m];
    acc = __builtin_amdgcn_wmma_f32_16x16x4_f32(false, a, false, b,
                                                (short)0, acc, false, false);
  }
  // D 16x16 f32: VGPR r -> M = r + 8*half, N = lane%16
  if (m0 + 16 <= N) {
    // uniform fast path: no per-lane guards, plain stores
#pragma unroll
    for (int r = 0; r < 8; ++r)
      h1[(m0 + r + (half << 3)) * HID + n0 + m] = acc[r];
  } else {
#pragma unroll
    for (int r = 0; r < 8; ++r) {
      int row = m0 + r + (half << 3);
      if (row < N) h1[row * HID + n0 + m] = acc[r];
    }
  }
}

// ---------------- g1 = b1 + h1 * dinv^2  (bias + self-loop message), float4 ----------------
__global__ void k_selfbias1(const float4* __restrict__ h1, const float* __restrict__ dinv,
                            const float4* __restrict__ b1v, float4* __restrict__ g1, int N) {
  int idx = blockIdx.x * blockDim.x + threadIdx.x;
  int total = N * (HID / 4);            // float4 elements
  if (idx < total) {
    int i  = idx >> 5;                  // 32 float4 per row
    int c4 = idx & 31;
    float w = dinv[i]; w *= w;
    float4 h = h1[idx];
    float4 b = b1v[c4];
    float4 r;
    r.x = b.x + h.x * w;
    r.y = b.y + h.y * w;
    r.z = b.z + h.z * w;
    r.w = b.w + h.w * w;
    g1[idx] = r;
  }
}

// ---------------- edge scatter-add, one wave32 per edge, float4 per lane ----------------
__global__ __launch_bounds__(256) void k_agg1(const int* __restrict__ ei,
                                              const float* __restrict__ dinv,
                                              const float* __restrict__ h1,
                                              float* __restrict__ g1, int E) {
  int wave = (int)((blockIdx.x * (unsigned)blockDim.x + threadIdx.x) >> 5);
  int lane = threadIdx.x & 31;
  if (wave >= E) return;
  int r = ei[wave];
  int c = ei[E + wave];
  float w = dinv[r] * dinv[c];
  const float4* src = (const float4*)(h1 + r * HID);
  float4 v = src[lane];
  float* dst = g1 + c * HID + lane * 4;
  atomicAdd(dst + 0, v.x * w);
  atomicAdd(dst + 1, v.y * w);
  atomicAdd(dst + 2, v.z * w);
  atomicAdd(dst + 3, v.w * w);
}

// ---------------- per-channel BN partial sums ----------------
__global__ void k_bnstats(const float* __restrict__ g1, float* __restrict__ sums,
                          float* __restrict__ sumsq, int N) {
  int c = threadIdx.x;                 // blockDim.x == 128
  float s = 0.f, s2 = 0.f;
  for (int i = blockIdx.x; i < N; i += gridDim.x) {
    float v = g1[i * HID + c];
    s += v; s2 += v * v;
  }
  atomicAdd(&sums[c], s);
  atomicAdd(&sumsq[c], s2);
}

// ---------------- fold BN into per-channel scale/shift ----------------
__global__ void k_bnfinal(const float* __restrict__ sums, const float* __restrict__ sumsq,
                          const float* __restrict__ gamma, const float* __restrict__ beta,
                          float* __restrict__ scale, float* __restrict__ shift, int N) {
  int c = threadIdx.x;
  if (c < HID) {
    float invN = 1.0f / (float)N;
    float mu  = sums[c] * invN;
    float var = sumsq[c] * invN - mu * mu;
    float a   = rsqrtf(var + BN_EPS) * gamma[c];
    scale[c] = a;
    shift[c] = beta[c] - mu * a;
  }
}

// ---------------- fused BN + ReLU + (128->2) GEMV; one wave32 per node ----------------
__global__ __launch_bounds__(256) void k_bngemv(const float* __restrict__ g1,
                                                const float* __restrict__ scale,
                                                const float* __restrict__ shift,
                                                const float* __restrict__ W2,
                                                float* __restrict__ t2, int N) {
  int node = (int)((blockIdx.x * (unsigned)blockDim.x + threadIdx.x) >> 5);
  int lane = threadIdx.x & 31;
  if (node >= N) return;
  const float* row = g1 + node * HID;
  float a0 = 0.f, a1 = 0.f;
#pragma unroll
  for (int j = 0; j < 4; ++j) {
    int c = lane + j * 32;
    float v = fmaxf(row[c] * scale[c] + shift[c], 0.0f);
    a0 += v * W2[c * OUTD + 0];
    a1 += v * W2[c * OUTD + 1];
  }
  for (int o = 16; o > 0; o >>= 1) {     // wave32 reduction
    a0 += __shfl_xor(a0, o, 32);
    a1 += __shfl_xor(a1, o, 32);
  }
  if (lane == 0) { t2[node * 2] = a0; t2[node * 2 + 1] = a1; }
}

// ---------------- out = b2 + t2 * dinv^2 (bias + self-loop) ----------------
__global__ void k_selfbias2(const float* __restrict__ t2, const float* __restrict__ dinv,
                            const float* __restrict__ b2, float* __restrict__ out, int N) {
  int i = blockIdx.x * blockDim.x + threadIdx.x;
  if (i < N) {
    float w = dinv[i]; w *= w;
    out[i * 2 + 0] = b2[0] + t2[i * 2 + 0] * w;
    out[i * 2 + 1] = b2[1] + t2[i * 2 + 1] * w;
  }
}

// ---------------- second-layer edge scatter (2 floats/edge) ----------------
__global__ void k_agg2(const int* __restrict__ ei, const float* __restrict__ dinv,
                       const float2* __restrict__ t2, float* __restrict__ out, int E) {
  int e = blockIdx.x * blockDim.x + threadIdx.x;
  if (e < E) {
    int r = ei[e], c = ei[E + e];
    float w = dinv[r] * dinv[c];
    float2 t = t2[r];
    atomicAdd(&out[c * 2 + 0], t.x * w);
    atomicAdd(&out[c * 2 + 1], t.y * w);
  }
}

extern "C" void kernel_launch(void* const* d_in, const int* in_sizes, int n_in,
                              void* d_out, int out_size, void* d_ws, size_t ws_size,
                              hipStream_t stream) {
  const float* x     = (const float*)d_in[0];
  const int*   ei    = (const int*)d_in[1];   // [2, E] flat: row then col
  const float* fi    = (const float*)d_in[2];
  const float* W1    = (const float*)d_in[3];
  const float* b1    = (const float*)d_in[4];
  const float* W2    = (const float*)d_in[5];
  const float* b2    = (const float*)d_in[6];
  const float* gamma = (const float*)d_in[7];
  const float* beta  = (const float*)d_in[8];
  float* out = (float*)d_out;

  const int N = in_sizes[0] / IN_DIM;
  const int E = in_sizes[1] / 2;

  // workspace layout (floats; all offsets 16B-aligned)
  float* ws = (float*)d_ws;
  size_t o = 0;
  float* sigw  = ws + o; o += 256;
  float* dinv  = ws + o; o += (size_t)((N + 3) & ~3);
  float* sums  = ws + o; o += 128;
  float* sumsq = ws + o; o += 128;
  float* scale = ws + o; o += 128;
  float* shift = ws + o; o += 128;
  float* h1    = ws + o; o += (size_t)N * HID;
  float* g1    = ws + o; o += (size_t)N * HID;
  float* t2    = ws + o; o += (size_t)N * 2;

  const int T = 256;
  k_init     <<<(N + T - 1) / T, T, 0, stream>>>(dinv, sums, sumsq, sigw, fi, N);
  k_deg      <<<(E + T - 1) / T, T, 0, stream>>>(ei, dinv, E);
  k_dinv     <<<(N + T - 1) / T, T, 0, stream>>>(dinv, N);
  k_gemm1    <<<(N + 15) / 16, 256, 0, stream>>>(x, W1, sigw, h1, N);
  {
    int total4 = N * (HID / 4);
    k_selfbias1<<<(total4 + T - 1) / T, T, 0, stream>>>((const float4*)h1, dinv,
                                                        (const float4*)b1, (float4*)g1, N);
  }
  {
    long long thr = (long long)E * 32;
    k_agg1<<<(unsigned)((thr + T - 1) / T), T, 0, stream>>>(ei, dinv, h1, g1, E);
  }
  k_bnstats  <<<1024, 128, 0, stream>>>(g1, sums, sumsq, N);
  k_bnfinal  <<<1, 128, 0, stream>>>(sums, sumsq, gamma, beta, scale, shift, N);
  {
    long long thr = (long long)N * 32;
    k_bngemv<<<(unsigned)((thr + T - 1) / T), T, 0, stream>>>(g1, scale, shift, W2, t2, N);
  }
  k_selfbias2<<<(N + T - 1) / T, T, 0, stream>>>(t2, dinv, b2, out, N);
  k_agg2     <<<(E + T - 1) / T, T, 0, stream>>>(ei, dinv, (const float2*)t2, out, E);
}